// GCNResnet_61486751809855
// MI455X (gfx1250) — compile-verified
//
#include <hip/hip_runtime.h>
#include <hip/hip_bf16.h>

// ---- problem sizes (match reference) ----
#define BB      64
#define CC      80
#define IN_CH   300
#define JJ      8192
#define SS      8
#define OUTD    1024          // J / S
#define FEAT    2048
#define HID     1024
#define SPAT    196           // 14*14

// ---- WMMA-GEMM tiling (TDM-staged) ----
#define NTB     128           // N columns per block (8 waves x 16)
#define KC      64            // K elements staged per LDS buffer (256B/row)

typedef __attribute__((ext_vector_type(2))) float v2f;
typedef __attribute__((ext_vector_type(8))) float v8f;
typedef __attribute__((ext_vector_type(4))) unsigned int v4u;
typedef __attribute__((ext_vector_type(8))) int v8i;
typedef __attribute__((ext_vector_type(4))) int v4i;

// ---------------------------------------------------------------------------
// TDM: issue one 2D tile DMA  global(row-major, row_stride=K elems) -> LDS
// (contiguous).  D# packing per CDNA5 ISA.
// tile = tile_rows x tile_k f32 elements.  6-arg builtin (clang-23 lane).
// ---------------------------------------------------------------------------
__device__ __forceinline__ void tdm_load_tile_f32(unsigned long long gaddr,
                                                  unsigned int lds_addr,
                                                  unsigned int tile_k,
                                                  unsigned int tile_rows,
                                                  unsigned int tensor_k,
                                                  unsigned long long row_stride) {
    // group0: [1:0]=count=1 | [63:32]=lds_addr | [120:64]=global_addr | [127:126]=type=2
    v4u g0 = { 1u,
               lds_addr,
               (unsigned int)gaddr,
               (unsigned int)((gaddr >> 32) & 0x01FFFFFFu) | (2u << 30) };
    // group1 dwords:
    // d0: wg_mask=0 | data_size(17:16)=2 (4B) | flags/pad = 0
    // d1: [15:0] atomic_barrier=0 | [31:16] tensor_dim0[15:0]
    // d2: [15:0] tensor_dim0[31:16] | [31:16] tensor_dim1[15:0]
    // d3: [15:0] tensor_dim1[31:16] | [31:16] tile_dim0
    // d4: [15:0] tile_dim1 | [31:16] tile_dim2=0
    // d5: tensor_dim0_stride[31:0]
    // d6: [15:0] tensor_dim0_stride[47:32] | [31:16] tensor_dim1_stride[15:0]=0
    // d7: tensor_dim1_stride[47:16]=0
    unsigned int td0 = tensor_k;
    unsigned int td1 = 0x7FFFFFFFu;        // huge: tiles never OOB in dim1
    v8i g1 = { (int)0x00020000u,
               (int)((td0 & 0xFFFFu) << 16),
               (int)((td0 >> 16) | ((td1 & 0xFFFFu) << 16)),
               (int)((td1 >> 16) | (tile_k << 16)),
               (int)(tile_rows & 0xFFFFu),
               (int)(unsigned int)row_stride,
               (int)((unsigned int)(row_stride >> 32) & 0xFFFFu),
               0 };
    v4i z4 = { 0, 0, 0, 0 };               // groups 2/3 unused (2D tensor)
    v8i z8 = { 0, 0, 0, 0, 0, 0, 0, 0 };
    __builtin_amdgcn_tensor_load_to_lds(g0, g1, z4, z4, z8, 0);
}

// ---------------------------------------------------------------------------
// K0: global spatial max over 14x14 -> feat[B, 2048]; one wave32 per (b,ch)
// ---------------------------------------------------------------------------
__global__ __launch_bounds__(256) void k_maxpool(const float* __restrict__ feature,
                                                 float* __restrict__ feat) {
    int gw   = (blockIdx.x * blockDim.x + threadIdx.x) >> 5;
    int lane = threadIdx.x & 31;
    if (gw >= BB * FEAT) return;
    const float* src = feature + (size_t)gw * SPAT;
    float m = -__builtin_huge_valf();
    for (int i = lane; i < SPAT; i += 32) m = fmaxf(m, src[i]);
    #pragma unroll
    for (int off = 16; off >= 1; off >>= 1) m = fmaxf(m, __shfl_xor(m, off, 32));
    if (lane == 0) feat[gw] = m;
}

// ---------------------------------------------------------------------------
// K1: adj = A.T * dinv[:,None] * dinv[None,:], dinv = rowsum(A)^-0.5
// ---------------------------------------------------------------------------
__global__ __launch_bounds__(128) void k_adj(const float* __restrict__ A,
                                             float* __restrict__ adj) {
    __shared__ float s_dinv[CC];
    int t = threadIdx.x;
    if (t < CC) {
        float s = 0.f;
        for (int j = 0; j < CC; ++j) s += A[t * CC + j];
        s_dinv[t] = 1.0f / sqrtf(s);
    }
    __syncthreads();
    for (int i = t; i < CC * CC; i += blockDim.x) {
        int r = i / CC, c = i % CC;
        adj[i] = A[c * CC + r] * s_dinv[r] * s_dinv[c];
    }
}

// ---------------------------------------------------------------------------
// K2: small row-major GEMM  C[M,N] = A[M,K] @ B[K,N]   (GCN projections)
// ---------------------------------------------------------------------------
__global__ __launch_bounds__(256) void k_gemm_rm(const float* __restrict__ A,
                                                 const float* __restrict__ B,
                                                 float* __restrict__ Cm,
                                                 int M, int N, int K) {
    int idx = blockIdx.x * blockDim.x + threadIdx.x;
    if (idx >= M * N) return;
    int m = idx / N, n = idx % N;
    const float* ap = A + (size_t)m * K;
    float acc = 0.f;
    for (int k = 0; k < K; ++k) acc = fmaf(ap[k], B[(size_t)k * N + n], acc);
    Cm[idx] = acc;
}

// ---------------------------------------------------------------------------
// K3: X[c,n] = act( sum_k adj[c,k] * T[k,n] ),  act = leaky_relu(0.2) or id
// ---------------------------------------------------------------------------
__global__ __launch_bounds__(256) void k_adjmul(const float* __restrict__ adj,
                                                const float* __restrict__ T,
                                                float* __restrict__ X,
                                                int N, int leaky) {
    int idx = blockIdx.x * blockDim.x + threadIdx.x;
    if (idx >= CC * N) return;
    int c = idx / N, n = idx % N;
    float acc = 0.f;
    for (int k = 0; k < CC; ++k) acc = fmaf(adj[c * CC + k], T[(size_t)k * N + n], acc);
    if (leaky) acc = acc > 0.f ? acc : 0.2f * acc;
    X[idx] = acc;
}

// ---------------------------------------------------------------------------
// K4/K5: WMMA f32 GEMM  C[M,N] = A[M,K] @ B[N,K]^T + bias[N]
// Block = 8 waves = 16(M) x 128(N) tile. The B (weight) stream — the
// dominant HBM traffic — is staged by the Tensor Data Mover into LDS,
// double-buffered (issue-ahead 1 chunk, s_wait_tensorcnt), and consumed as
// per-lane float2 fragments via ds_load_b64.  A (tiny, L2-resident) is read
// directly from global.  One v_wmma_f32_16x16x4_f32 per K=4 step.
// ---------------------------------------------------------------------------
__global__ __launch_bounds__(256) void k_wmma_abT_bias(const float* __restrict__ A,
                                                       const float* __restrict__ B,
                                                       const float* __restrict__ bias,
                                                       float* __restrict__ C,
                                                       int M, int N, int K) {
    __shared__ float ldsb[2][NTB * KC];       // 2 x 32 KB

    int wv     = threadIdx.x >> 5;            // 0..7 : n-subtile within block
    int lane   = threadIdx.x & 31;
    int mtiles = M >> 4;
    int mt     = blockIdx.x % mtiles;         // same-nb blocks adjacent -> L2 reuse
    int nb     = blockIdx.x / mtiles;

    int row  = lane & 15;                     // M (A) / N (B) index
    int koff = (lane >> 4) << 1;              // 0 or 2 (K half-pair)
    const float* ap = A + (size_t)(mt * 16 + row) * K + koff;

    int nchunks = K / KC;
    unsigned long long gbase =
        (unsigned long long)(uintptr_t)(B + (size_t)nb * NTB * K);
    // LDS byte offset = low 32 bits of flat shared address (ISA §10.2)
    unsigned int lds0 = (unsigned int)(uintptr_t)(const void*)&ldsb[0][0];
    unsigned int lds1 = (unsigned int)(uintptr_t)(const void*)&ldsb[1][0];

    if (threadIdx.x < 32)                     // one TDM issue per block
        tdm_load_tile_f32(gbase, lds0, KC, NTB, (unsigned)K,
                          (unsigned long long)(unsigned)K);

    v8f acc = {};
    for (int c = 0; c < nchunks; ++c) {
        if (threadIdx.x < 32) {
            if (c + 1 < nchunks) {
                tdm_load_tile_f32(gbase + (unsigned long long)(c + 1) * KC * 4u,
                                  ((c + 1) & 1) ? lds1 : lds0,
                                  KC, NTB, (unsigned)K,
                                  (unsigned long long)(unsigned)K);
                __builtin_amdgcn_s_wait_tensorcnt(1);   // chunk c complete
            } else {
                __builtin_amdgcn_s_wait_tensorcnt(0);
            }
        }
        __syncthreads();                      // chunk c visible to all waves

        const float* bp = &ldsb[c & 1][(wv * 16 + row) * KC + koff];
        const float* ac = ap + c * KC;
        #pragma unroll
        for (int k0 = 0; k0 < KC; k0 += 4) {
            v2f a = *(const v2f*)(ac + k0);
            v2f b = *(const v2f*)(bp + k0);   // ds_load_b64
            acc = __builtin_amdgcn_wmma_f32_16x16x4_f32(false, a, false, b,
                                                        (short)0, acc, false, false);
        }
        __syncthreads();                      // buffer reusable next iteration
    }

    // D layout: VGPR r -> (M = mt*16 + 8*(lane>>4) + r, N = lane&15 of subtile)
    int nidx  = nb * NTB + wv * 16 + (lane & 15);
    float bv  = bias[nidx];
    int mbase = mt * 16 + ((lane >> 4) << 3);
    #pragma unroll
    for (int r = 0; r < 8; ++r)
        C[(size_t)(mbase + r) * N + nidx] = acc[r] + bv;
}

// ---------------------------------------------------------------------------
// K6: MFB sum-pool  pooled[b,c,o] = sum_s img[b,o*8+s] * cls[c,o*8+s]
// img (2 MB) and cls (2.5 MB) are L2-resident; coalesced float4 loads.
// ---------------------------------------------------------------------------
__global__ __launch_bounds__(256) void k_pooled(const float* __restrict__ img,
                                                const float* __restrict__ cls,
                                                float* __restrict__ pooled) {
    int idx = blockIdx.x * blockDim.x + threadIdx.x;
    if (idx >= BB * CC * OUTD) return;
    int o = idx & (OUTD - 1);
    int c = (idx >> 10) % CC;
    int b = idx / (CC * OUTD);
    const float4* ip = (const float4*)(img + (size_t)b * JJ + o * SS);
    const float4* cp = (const float4*)(cls + (size_t)c * JJ + o * SS);
    float4 i0 = ip[0], i1 = ip[1], c0 = cp[0], c1 = cp[1];
    pooled[idx] = i0.x*c0.x + i0.y*c0.y + i0.z*c0.z + i0.w*c0.w
                + i1.x*c1.x + i1.y*c1.y + i1.z*c1.z + i1.w*c1.w;
}

// ---------------------------------------------------------------------------
// K7a: out[b,c] = b_ml[c]  (accumulation base for atomic K-split GEMM)
// ---------------------------------------------------------------------------
__global__ __launch_bounds__(256) void k_out_init(const float* __restrict__ b_ml,
                                                  float* __restrict__ out) {
    int idx = blockIdx.x * blockDim.x + threadIdx.x;
    if (idx < BB * CC) out[idx] = b_ml[idx % CC];
}

// ---------------------------------------------------------------------------
// K7b: out += pooled[64, 81920] @ W_ml[80, 81920]^T, K split into chunks.
// One wave handles one (n-tile, k-chunk) and ALL 4 m-tiles, reusing the
// B fragment 4x; partial sums land via global_atomic_add_f32.
// ---------------------------------------------------------------------------
__global__ __launch_bounds__(256) void k_wmma_final(const float* __restrict__ A,
                                                    const float* __restrict__ B,
                                                    float* __restrict__ C,
                                                    int N, int K, int kchunk) {
    int gw   = (blockIdx.x * blockDim.x + threadIdx.x) >> 5;
    int lane = threadIdx.x & 31;
    int nkc  = K / kchunk;
    int nt   = gw / nkc;
    int kc   = gw % nkc;
    if (nt >= (N >> 4)) return;                 // wave-uniform exit

    int row  = lane & 15;
    int koff = (lane >> 4) << 1;
    const float* bp  = B + (size_t)(nt * 16 + row) * K + (size_t)kc * kchunk + koff;
    const float* ap0 = A + (size_t)row * K + (size_t)kc * kchunk + koff;

    v8f acc0 = {}, acc1 = {}, acc2 = {}, acc3 = {};
    #pragma unroll 4
    for (int k0 = 0; k0 < kchunk; k0 += 4) {
        v2f b  = *(const v2f*)(bp + k0);
        v2f a0 = *(const v2f*)(ap0 + k0);
        v2f a1 = *(const v2f*)(ap0 + (size_t)16 * K + k0);
        v2f a2 = *(const v2f*)(ap0 + (size_t)32 * K + k0);
        v2f a3 = *(const v2f*)(ap0 + (size_t)48 * K + k0);
        acc0 = __builtin_amdgcn_wmma_f32_16x16x4_f32(false, a0, false, b, (short)0, acc0, false, false);
        acc1 = __builtin_amdgcn_wmma_f32_16x16x4_f32(false, a1, false, b, (short)0, acc1, false, false);
        acc2 = __builtin_amdgcn_wmma_f32_16x16x4_f32(false, a2, false, b, (short)0, acc2, false, false);
        acc3 = __builtin_amdgcn_wmma_f32_16x16x4_f32(false, a3, false, b, (short)0, acc3, false, false);
    }

    int nidx  = nt * 16 + (lane & 15);
    int mbase = (lane >> 4) << 3;
    #pragma unroll
    for (int r = 0; r < 8; ++r) {
        atomicAdd(&C[(size_t)(mbase +  0 + r) * N + nidx], acc0[r]);
        atomicAdd(&C[(size_t)(mbase + 16 + r) * N + nidx], acc1[r]);
        atomicAdd(&C[(size_t)(mbase + 32 + r) * N + nidx], acc2[r]);
        atomicAdd(&C[(size_t)(mbase + 48 + r) * N + nidx], acc3[r]);
    }
}

// ---------------------------------------------------------------------------
extern "C" void kernel_launch(void* const* d_in, const int* in_sizes, int n_in,
                              void* d_out, int out_size, void* d_ws, size_t ws_size,
                              hipStream_t stream) {
    (void)in_sizes; (void)n_in; (void)out_size; (void)ws_size;
    const float* feature = (const float*)d_in[0];   // [64,2048,14,14]
    const float* inp     = (const float*)d_in[1];   // [1,80,300]
    const float* Aco     = (const float*)d_in[2];   // [80,80]
    const float* W_gc1   = (const float*)d_in[3];   // [300,1024]
    const float* W_gc2   = (const float*)d_in[4];   // [1024,2048]
    const float* W_img   = (const float*)d_in[5];   // [8192,2048]
    const float* b_img   = (const float*)d_in[6];   // [8192]
    const float* W_cls   = (const float*)d_in[7];   // [8192,2048]
    const float* b_cls   = (const float*)d_in[8];   // [8192]
    const float* W_ml    = (const float*)d_in[9];   // [80,81920]
    const float* b_ml    = (const float*)d_in[10];  // [80]
    float* out = (float*)d_out;                     // [64,80]
    float* ws  = (float*)d_ws;

    // workspace layout (floats)
    float* feat   = ws;                 // 131072
    float* adj    = feat   + BB*FEAT;   // 6400
    float* t1     = adj    + CC*CC;     // 81920
    float* x1     = t1     + CC*HID;    // 81920
    float* t2     = x1     + CC*HID;    // 163840
    float* x2     = t2     + CC*FEAT;   // 163840
    float* img    = x2     + CC*FEAT;   // 524288
    float* cls    = img    + BB*JJ;     // 655360
    float* pooled = cls    + CC*JJ;     // 5242880  (~28.2 MB total)

    // K0: max-pool  (64*2048 waves)
    k_maxpool<<<(BB*FEAT*32 + 255)/256, 256, 0, stream>>>(feature, feat);

    // K1: normalized adjacency
    k_adj<<<1, 128, 0, stream>>>(Aco, adj);

    // K2/K3: GCN layer 1
    k_gemm_rm<<<(CC*HID + 255)/256, 256, 0, stream>>>(inp, W_gc1, t1, CC, HID, IN_CH);
    k_adjmul<<<(CC*HID + 255)/256, 256, 0, stream>>>(adj, t1, x1, HID, 1);
    // GCN layer 2
    k_gemm_rm<<<(CC*FEAT + 255)/256, 256, 0, stream>>>(x1, W_gc2, t2, CC, FEAT, HID);
    k_adjmul<<<(CC*FEAT + 255)/256, 256, 0, stream>>>(adj, t2, x2, FEAT, 0);

    // K4: img = feat @ W_img^T + b_img   (4 mtiles x 64 nblocks = 256 blocks)
    k_wmma_abT_bias<<<(BB/16) * (JJ/NTB), 256, 0, stream>>>(
        feat, W_img, b_img, img, BB, JJ, FEAT);
    // K5: cls = x2 @ W_cls^T + b_cls     (5 mtiles x 64 nblocks = 320 blocks)
    k_wmma_abT_bias<<<(CC/16) * (JJ/NTB), 256, 0, stream>>>(
        x2, W_cls, b_cls, cls, CC, JJ, FEAT);

    // K6: MFB sum-pool -> pooled [64,80,1024]
    k_pooled<<<(BB*CC*OUTD + 255)/256, 256, 0, stream>>>(img, cls, pooled);

    // K7: out = pooled_flat @ W_ml^T + b_ml   (K = 81920 split into 512-chunks)
    k_out_init<<<(BB*CC + 255)/256, 256, 0, stream>>>(b_ml, out);
    {
        const int KTOT = CC * OUTD;     // 81920
        const int kchunk = 512;         // 160 chunks -> 5*160 = 800 waves
        int waves = (CC/16) * (KTOT / kchunk);
        k_wmma_final<<<(waves*32 + 255)/256, 256, 0, stream>>>(
            pooled, W_ml, out, CC, KTOT, kchunk);
    }
}